// GATConvBlock_19954418057458
// MI455X (gfx1250) — compile-verified
//
#include <hip/hip_runtime.h>

typedef __attribute__((ext_vector_type(2))) float v2f;
typedef __attribute__((ext_vector_type(8))) float v8f;

#define N_NODES 100000
#define N_EDGES 1600000
#define NEG_SLOPE 0.2f
#define LN_EPS 1e-5f

// ---------------- helpers ----------------
__device__ __forceinline__ float leaky(float v) { return v > 0.0f ? v : NEG_SLOPE * v; }

__device__ __forceinline__ unsigned f2ord(float f) {
    unsigned u = __float_as_uint(f);
    return (u >> 31) ? ~u : (u | 0x80000000u);
}
__device__ __forceinline__ float ord2f(unsigned u) {
    if (u == 0u) return 0.0f;  // "no edge" sentinel (never hit: self-loops)
    return (u >> 31) ? __uint_as_float(u & 0x7fffffffu) : __uint_as_float(~u);
}

__device__ __forceinline__ float wave_sum(float v) {
    #pragma unroll
    for (int off = 16; off; off >>= 1) v += __shfl_xor(v, off, 32);
    return v;
}

// ---------------- zero scratch ----------------
__global__ void zero_kernel(float* __restrict__ p, long n) {
    long i = (long)blockIdx.x * blockDim.x + threadIdx.x;
    if (i < n) p[i] = 0.0f;
}

// ---------------- GEMM: H = X @ W via V_WMMA_F32_16X16X4_F32 ----------------
// Block = 256 threads = 8 waves. Each block: one 16-row tile; wave w: 16-col tile w.
__global__ void __launch_bounds__(256) gemm_wmma_kernel(
    const float* __restrict__ X, const float* __restrict__ W, float* __restrict__ H) {
    const int lane   = threadIdx.x & 31;
    const int wave   = threadIdx.x >> 5;       // 0..7 -> column tile
    const int rowBase = blockIdx.x * 16;       // 100000 = 6250*16, no tail
    const int colBase = wave * 16;
    const int m       = lane & 15;
    const int halfsel = lane >> 4;             // 0: K=0,1  1: K=2,3
    const int kb      = halfsel * 2;

    const float* xrow = X + (size_t)(rowBase + m) * 128;
    const float* wcol = W + colBase + (lane & 15);

    v8f acc = {};
    #pragma unroll 4
    for (int kt = 0; kt < 128; kt += 4) {
        v2f a, b;
        const float2 av = *(const float2*)(xrow + kt + kb);   // contiguous K pair
        a.x = av.x; a.y = av.y;
        b.x = wcol[(size_t)(kt + kb + 0) * 128];
        b.y = wcol[(size_t)(kt + kb + 1) * 128];
        acc = __builtin_amdgcn_wmma_f32_16x16x4_f32(
            /*neg_a=*/false, a, /*neg_b=*/false, b,
            /*c_mod=*/(short)0, acc, /*reuse_a=*/false, /*reuse_b=*/false);
    }

    // D layout: VGPR r -> row r (lanes 0-15) / row r+8 (lanes 16-31), col = lane&15
    float* out = H + (size_t)rowBase * 128 + colBase + (lane & 15);
    const int rOff = halfsel * 8;
    #pragma unroll
    for (int r = 0; r < 8; ++r)
        out[(size_t)(r + rOff) * 128] = acc[r];
}

// ---------------- per-node attention coefficients ----------------
// One wave per node; lane covers channels 4l..4l+3 (head = lane>>3).
__global__ void __launch_bounds__(256) attn_coef_kernel(
    const float* __restrict__ H, const float* __restrict__ att_src,
    const float* __restrict__ att_dst, float* __restrict__ aS, float* __restrict__ aD) {
    const int gid  = blockIdx.x * blockDim.x + threadIdx.x;
    const int node = gid >> 5;
    const int lane = gid & 31;
    if (node >= N_NODES) return;

    const float4 hv = *(const float4*)(H + (size_t)node * 128 + lane * 4);
    const float4 as = *(const float4*)(att_src + lane * 4);  // flat idx == channel idx
    const float4 ad = *(const float4*)(att_dst + lane * 4);
    float ps = hv.x * as.x + hv.y * as.y + hv.z * as.z + hv.w * as.w;
    float pd = hv.x * ad.x + hv.y * ad.y + hv.z * ad.z + hv.w * ad.w;
    #pragma unroll
    for (int off = 4; off; off >>= 1) {                      // reduce within 8-lane head group
        ps += __shfl_xor(ps, off, 32);
        pd += __shfl_xor(pd, off, 32);
    }
    if ((lane & 7) == 0) {
        const int head = lane >> 3;
        aS[(size_t)node * 4 + head] = ps;
        aD[(size_t)node * 4 + head] = pd;
    }
}

// ---------------- edge pass 1: segment max via ordered-uint atomicMax ----------------
__global__ void __launch_bounds__(256) edge_max_kernel(
    const int* __restrict__ ei, const float* __restrict__ aS,
    const float* __restrict__ aD, unsigned* __restrict__ keys) {
    const int e = blockIdx.x * blockDim.x + threadIdx.x;
    if (e >= N_EDGES + N_NODES) return;
    const int src = (e < N_EDGES) ? ei[e] : (e - N_EDGES);
    const int dst = (e < N_EDGES) ? ei[N_EDGES + e] : (e - N_EDGES);
    const float4 as = *(const float4*)(aS + (size_t)src * 4);
    const float4 ad = *(const float4*)(aD + (size_t)dst * 4);
    unsigned* kp = keys + (size_t)dst * 4;
    atomicMax(kp + 0, f2ord(leaky(as.x + ad.x)));
    atomicMax(kp + 1, f2ord(leaky(as.y + ad.y)));
    atomicMax(kp + 2, f2ord(leaky(as.z + ad.z)));
    atomicMax(kp + 3, f2ord(leaky(as.w + ad.w)));
}

__global__ void key2float_kernel(unsigned* __restrict__ keys, float* __restrict__ amax) {
    const int i = blockIdx.x * blockDim.x + threadIdx.x;
    if (i < N_NODES * 4) amax[i] = ord2f(keys[i]);   // in-place (same buffer)
}

// ---------------- edge pass 2: softmax denominator ----------------
__global__ void __launch_bounds__(256) edge_denom_kernel(
    const int* __restrict__ ei, const float* __restrict__ aS, const float* __restrict__ aD,
    const float* __restrict__ amax, float* __restrict__ denom) {
    const int e = blockIdx.x * blockDim.x + threadIdx.x;
    if (e >= N_EDGES + N_NODES) return;
    const int src = (e < N_EDGES) ? ei[e] : (e - N_EDGES);
    const int dst = (e < N_EDGES) ? ei[N_EDGES + e] : (e - N_EDGES);
    const float4 as = *(const float4*)(aS + (size_t)src * 4);
    const float4 ad = *(const float4*)(aD + (size_t)dst * 4);
    const float4 mx = *(const float4*)(amax + (size_t)dst * 4);
    float* dp = denom + (size_t)dst * 4;
    atomicAdd(dp + 0, __expf(leaky(as.x + ad.x) - mx.x));
    atomicAdd(dp + 1, __expf(leaky(as.y + ad.y) - mx.y));
    atomicAdd(dp + 2, __expf(leaky(as.z + ad.z) - mx.z));
    atomicAdd(dp + 3, __expf(leaky(as.w + ad.w) - mx.w));
}

// ---------------- edge pass 3: weighted scatter-add (bottleneck) ----------------
// One wave per edge; lane covers channels 4l..4l+3 of one head (head = lane>>3).
__global__ void __launch_bounds__(256) edge_agg_kernel(
    const int* __restrict__ ei, const float* __restrict__ H,
    const float* __restrict__ aS, const float* __restrict__ aD,
    const float* __restrict__ amax, const float* __restrict__ denom,
    float* __restrict__ accum) {
    const long gid  = (long)blockIdx.x * blockDim.x + threadIdx.x;
    const long e    = gid >> 5;
    const int  lane = (int)(gid & 31);
    if (e >= (long)(N_EDGES + N_NODES)) return;
    const int src = (e < N_EDGES) ? ei[e] : (int)(e - N_EDGES);
    const int dst = (e < N_EDGES) ? ei[N_EDGES + e] : (int)(e - N_EDGES);
    const int head = lane >> 3;

    const float a  = leaky(aS[(size_t)src * 4 + head] + aD[(size_t)dst * 4 + head]);
    const float mx = amax[(size_t)dst * 4 + head];
    const float dn = denom[(size_t)dst * 4 + head];
    const float w  = __expf(a - mx) / (dn + 1e-16f);

    const float4 hv = *(const float4*)(H + (size_t)src * 128 + lane * 4);
    float* op = accum + (size_t)dst * 128 + lane * 4;
    atomicAdd(op + 0, hv.x * w);
    atomicAdd(op + 1, hv.y * w);
    atomicAdd(op + 2, hv.z * w);
    atomicAdd(op + 3, hv.w * w);
}

// ---------------- finalize: +bias, LayerNorm(128), LeakyReLU ----------------
__global__ void __launch_bounds__(256) finalize_kernel(
    const float* __restrict__ accum, const float* __restrict__ bias,
    const float* __restrict__ ln_w, const float* __restrict__ ln_b,
    float* __restrict__ out) {
    const int gid  = blockIdx.x * blockDim.x + threadIdx.x;
    const int node = gid >> 5;
    const int lane = gid & 31;
    if (node >= N_NODES) return;

    float4 v = *(const float4*)(accum + (size_t)node * 128 + lane * 4);
    const float4 bb = *(const float4*)(bias + lane * 4);
    v.x += bb.x; v.y += bb.y; v.z += bb.z; v.w += bb.w;

    const float mu = wave_sum(v.x + v.y + v.z + v.w) * (1.0f / 128.0f);
    const float dx = v.x - mu, dy = v.y - mu, dz = v.z - mu, dw = v.w - mu;
    const float var = wave_sum(dx * dx + dy * dy + dz * dz + dw * dw) * (1.0f / 128.0f);
    const float inv = rsqrtf(var + LN_EPS);

    const float4 gw = *(const float4*)(ln_w + lane * 4);
    const float4 gb = *(const float4*)(ln_b + lane * 4);
    float4 y;
    y.x = leaky(dx * inv * gw.x + gb.x);
    y.y = leaky(dy * inv * gw.y + gb.y);
    y.z = leaky(dz * inv * gw.z + gb.z);
    y.w = leaky(dw * inv * gw.w + gb.w);
    *(float4*)(out + (size_t)node * 128 + lane * 4) = y;
}

// ---------------- launcher ----------------
extern "C" void kernel_launch(void* const* d_in, const int* in_sizes, int n_in,
                              void* d_out, int out_size, void* d_ws, size_t ws_size,
                              hipStream_t stream) {
    const float* x       = (const float*)d_in[0];
    const int*   ei      = (const int*)  d_in[1];
    const float* W       = (const float*)d_in[2];
    const float* att_src = (const float*)d_in[3];
    const float* att_dst = (const float*)d_in[4];
    const float* bias    = (const float*)d_in[5];
    const float* ln_w    = (const float*)d_in[6];
    const float* ln_b    = (const float*)d_in[7];
    float* out = (float*)d_out;

    const size_t N = N_NODES;
    char* ws = (char*)d_ws;
    float* H     = (float*)ws;                       // N*128
    float* aS    = (float*)(ws + N * 128 * 4);       // N*4
    float* aD    = aS + N * 4;                       // N*4
    float* amax  = aD + N * 4;                       // N*4 (uint keys, then floats)
    float* denom = amax + N * 4;                     // N*4
    float* accum = denom + N * 4;                    // N*128

    const int EP = N_EDGES + N_NODES;                // edges + self loops

    // zero amax+denom+accum (contiguous)
    const long zn = (long)N * 4 * 2 + (long)N * 128;
    zero_kernel<<<(int)((zn + 255) / 256), 256, 0, stream>>>(amax, zn);

    gemm_wmma_kernel<<<N_NODES / 16, 256, 0, stream>>>(x, W, H);

    attn_coef_kernel<<<(N_NODES * 32 + 255) / 256, 256, 0, stream>>>(H, att_src, att_dst, aS, aD);

    edge_max_kernel<<<(EP + 255) / 256, 256, 0, stream>>>(ei, aS, aD, (unsigned*)amax);

    key2float_kernel<<<(N_NODES * 4 + 255) / 256, 256, 0, stream>>>((unsigned*)amax, amax);

    edge_denom_kernel<<<(EP + 255) / 256, 256, 0, stream>>>(ei, aS, aD, amax, denom);

    edge_agg_kernel<<<(int)(((long)EP * 32 + 255) / 256), 256, 0, stream>>>(
        ei, H, aS, aD, amax, denom, accum);

    finalize_kernel<<<(N_NODES * 32 + 255) / 256, 256, 0, stream>>>(accum, bias, ln_w, ln_b, out);
}